// TorchNeighborList_1614907703797
// MI455X (gfx1250) — compile-verified
//
#include <hip/hip_runtime.h>
#include <hip/hip_bf16.h>

typedef float v2f __attribute__((ext_vector_type(2)));
typedef float v8f __attribute__((ext_vector_type(8)));

#define N_ATOMS 4096
#define NTILE   (N_ATOMS / 16)
#define CUTOFF2 25.0f                       // 5.0^2
#define TWO_P   (N_ATOMS * (N_ATOMS - 1))   // 16,773,120 bidirectional entries

// ---------------------------------------------------------------------------
// wave32 ballot fed with a raw compare -> folds to the compare's lane mask
// ---------------------------------------------------------------------------
__device__ __forceinline__ unsigned ballot32(bool p) {
#if __has_builtin(__builtin_amdgcn_ballot_w32)
    return __builtin_amdgcn_ballot_w32(p);
#else
    return (unsigned)__ballot(p);
#endif
}

// ---------------------------------------------------------------------------
// Stage all 4096 positions (48 KB) into LDS with async global->LDS DMA copies
// (ASYNCcnt-tracked, ISA 15.18.3 GLOBAL_LOAD_ASYNC_TO_LDS_B128, GVS mode),
// then drain ASYNCcnt and barrier.  12288 floats = 3072 x 16B transfers.
// ---------------------------------------------------------------------------
__device__ __forceinline__ void stage_positions(const float* __restrict__ pos,
                                                float* sp) {
    for (int t = threadIdx.x; t < (N_ATOMS * 3) / 4; t += 256) {
        unsigned lds_addr = (unsigned)(size_t)(&sp[4 * t]); // low 32b = LDS offset
        unsigned goff     = (unsigned)t * 16u;
        asm volatile("global_load_async_to_lds_b128 %0, %1, %2"
                     :: "v"(lds_addr), "v"(goff), "s"(pos)
                     : "memory");
    }
    asm volatile("s_wait_asynccnt 0" ::: "memory");
    __syncthreads();
}

// ---------------------------------------------------------------------------
// WMMA tile algebra (no cross-lane broadcasts needed):
//   A[m][:] = (-2x_i, -2y_i, -2z_i, r2_i - C2)   (16x4 f32)
//   B[:][n] = (  x_j,   y_j,   z_j, 1 )
//   D[m][n] = -2 p_i.p_j + r2_i - C2
//   in-cutoff  <=>  D[m][n] < -r2_j   (RHS is a per-lane/column scalar)
// Diagonal-tile constraints applied as 16-bit column masks on the ballot.
// ---------------------------------------------------------------------------

// Kernel 1: per-atom neighbor counts.  One wave per I-tile; the wave owns its
// 16 rows exclusively -> plain stores, no atomics, fully deterministic.
__global__ __launch_bounds__(256) void nl_count_wmma(const float* __restrict__ pos,
                                                     int* __restrict__ counts) {
    __shared__ alignas(16) float sp[N_ATOMS * 3];
    stage_positions(pos, sp);

    const int wave = threadIdx.x >> 5;
    const int lane = threadIdx.x & 31;
    const int m    = lane & 15;
    const int hi   = lane >> 4;
    const int tI   = blockIdx.x * 8 + wave;

    const int   ai  = tI * 16 + m;
    const float xi  = sp[3 * ai + 0], yi = sp[3 * ai + 1], zi = sp[3 * ai + 2];
    const float r2i = xi * xi + yi * yi + zi * zi;

    const bool hi0 = (hi == 0);
    v2f a;                                   // branchless: EXEC stays all-ones
    a.x = hi0 ? -2.0f * xi : -2.0f * zi;
    a.y = hi0 ? -2.0f * yi : (r2i - CUTOFF2);

    int cnt_r[8];
#pragma unroll
    for (int r = 0; r < 8; ++r) cnt_r[r] = 0;

    for (int s = 0; s < NTILE; ++s) {
        int tJ = tI + s; if (tJ >= NTILE) tJ -= NTILE;
        const int   aj  = tJ * 16 + m;
        const float xj  = sp[3 * aj + 0], yj = sp[3 * aj + 1], zj = sp[3 * aj + 2];
        const float negr2j = -(xj * xj + yj * yj + zj * zj);
        v2f b;
        b.x = hi0 ? xj : zj;
        b.y = hi0 ? yj : 1.0f;

        v8f c = {};
        v8f d = __builtin_amdgcn_wmma_f32_16x16x4_f32(
            false, a, false, b, (short)0, c, false, false);

        const bool diag = (s == 0);
#pragma unroll
        for (int r = 0; r < 8; ++r) {
            const int row = r + 8 * hi;
            // raw compare only -> ballot folds to the v_cmp lane mask
            unsigned mk   = ballot32(d[r] < negr2j);
            unsigned half = hi ? (mk >> 16) : (mk & 0xffffu);
            unsigned tmask = 0xffffu;
            if (diag) tmask = 0xffffu & ~(1u << row);   // exclude j == v (peeled iter)
            cnt_r[r] += __popc(half & tmask);
        }
    }
#pragma unroll
    for (int r = 0; r < 8; ++r)
        if (m == 0) counts[tI * 16 + 8 * hi + r] = cnt_r[r];
}

// ---------------------------------------------------------------------------
// Kernel 2: exclusive scan of 4096 counts -> offsets[4097]; writes num_pairs.
// ---------------------------------------------------------------------------
__global__ __launch_bounds__(1024) void nl_scan(const int* __restrict__ counts,
                                                int* __restrict__ offsets,
                                                int* __restrict__ npairs_out) {
    __shared__ int sums[1024];
    const int t  = threadIdx.x;
    const int c0 = counts[4 * t + 0];
    const int c1 = counts[4 * t + 1];
    const int c2 = counts[4 * t + 2];
    const int c3 = counts[4 * t + 3];
    sums[t] = c0 + c1 + c2 + c3;
    __syncthreads();
    for (int off = 1; off < 1024; off <<= 1) {
        int v = (t >= off) ? sums[t - off] : 0;
        __syncthreads();
        sums[t] += v;
        __syncthreads();
    }
    const int base = (t > 0) ? sums[t - 1] : 0;
    offsets[4 * t + 0] = base;
    offsets[4 * t + 1] = base + c0;
    offsets[4 * t + 2] = base + c0 + c1;
    offsets[4 * t + 3] = base + c0 + c1 + c2;
    if (t == 1023) {
        offsets[N_ATOMS] = sums[1023];
        *npairs_out      = sums[1023];
    }
}

// ---------------------------------------------------------------------------
// Kernel 3: fill all 2P slots with the reference's padding pattern
// (idx = N sentinel, Rij = 0, cell_offset = 0, mask = 0).
// ---------------------------------------------------------------------------
__global__ void nl_pad(int* __restrict__ idx_i, int* __restrict__ idx_j,
                       float* __restrict__ rij, int* __restrict__ cell,
                       int* __restrict__ mask) {
    const int k = blockIdx.x * blockDim.x + threadIdx.x;
    if (k >= TWO_P) return;
    idx_i[k] = N_ATOMS;
    idx_j[k] = N_ATOMS;
    mask[k]  = 0;
    rij[3 * k + 0] = 0.0f; rij[3 * k + 1] = 0.0f; rij[3 * k + 2] = 0.0f;
    cell[3 * k + 0] = 0;   cell[3 * k + 1] = 0;   cell[3 * k + 2] = 0;
}

// ---------------------------------------------------------------------------
// Kernel 4: WMMA-tiled emission in the reference's stable-sorted order.
// One wave per I-tile.  J-tiles walked as tJ = tI (cols j>v), tI+1 ... tI-1
// (all cols), tI (cols j<v): exactly "w = v+1..N-1 then 0..v-1" per row.
// Column masks on the ballot encode the diagonal constraints; ballot+popcount
// rank compacts columns in ascending order; per-row running offsets are
// uniform across each 16-lane half.
// ---------------------------------------------------------------------------
__global__ __launch_bounds__(256) void nl_emit_wmma(const float* __restrict__ pos,
                                                    const int* __restrict__ offsets,
                                                    int* __restrict__ idx_i,
                                                    int* __restrict__ idx_j,
                                                    float* __restrict__ rij,
                                                    int* __restrict__ mask_out) {
    __shared__ alignas(16) float sp[N_ATOMS * 3];
    stage_positions(pos, sp);

    const int wave = threadIdx.x >> 5;
    const int lane = threadIdx.x & 31;
    const int m    = lane & 15;
    const int hi   = lane >> 4;
    const int tI   = blockIdx.x * 8 + wave;

    const int   ai  = tI * 16 + m;
    const float xi  = sp[3 * ai + 0], yi = sp[3 * ai + 1], zi = sp[3 * ai + 2];
    const float r2i = xi * xi + yi * yi + zi * zi;

    const bool hi0 = (hi == 0);
    v2f a;
    a.x = hi0 ? -2.0f * xi : -2.0f * zi;
    a.y = hi0 ? -2.0f * yi : (r2i - CUTOFF2);

    int ofs[8];
#pragma unroll
    for (int r = 0; r < 8; ++r)
        ofs[r] = offsets[tI * 16 + 8 * hi + r];

    for (int s = 0; s <= NTILE; ++s) {
        int tJ = tI + s; if (tJ >= NTILE) tJ -= NTILE;
        const int   aj  = tJ * 16 + m;
        const float xj  = sp[3 * aj + 0], yj = sp[3 * aj + 1], zj = sp[3 * aj + 2];
        const float negr2j = -(xj * xj + yj * yj + zj * zj);
        v2f b;
        b.x = hi0 ? xj : zj;
        b.y = hi0 ? yj : 1.0f;

        v8f c = {};
        v8f d = __builtin_amdgcn_wmma_f32_16x16x4_f32(
            false, a, false, b, (short)0, c, false, false);

        const bool diagF = (s == 0);
        const bool diagL = (s == NTILE);
#pragma unroll
        for (int r = 0; r < 8; ++r) {
            const int row = r + 8 * hi;
            const int v   = tI * 16 + row;
            // raw compare only -> ballot folds to the v_cmp lane mask
            unsigned mk   = ballot32(d[r] < negr2j);
            unsigned half = hi ? (mk >> 16) : (mk & 0xffffu);
            unsigned tmask = 0xffffu;
            if (diagF) tmask = (0xfffeu << row) & 0xffffu;  // cols j > v (peeled)
            if (diagL) tmask = (1u << row) - 1u;            // cols j < v (peeled)
            const unsigned hmk = half & tmask;
            if ((hmk >> m) & 1u) {                          // this lane's column hit
                const int j    = tJ * 16 + m;
                const int rank = __popc(hmk & ((1u << m) - 1u));
                const int o    = ofs[r] + rank;
                idx_i[o]    = v;
                idx_j[o]    = j;
                mask_out[o] = 1;
                rij[3 * o + 0] = xj - sp[3 * v + 0];
                rij[3 * o + 1] = yj - sp[3 * v + 1];
                rij[3 * o + 2] = zj - sp[3 * v + 2];
            }
            ofs[r] += __popc(hmk);
        }
    }
}

// ---------------------------------------------------------------------------
extern "C" void kernel_launch(void* const* d_in, const int* in_sizes, int n_in,
                              void* d_out, int out_size, void* d_ws, size_t ws_size,
                              hipStream_t stream) {
    (void)in_sizes; (void)n_in; (void)out_size; (void)ws_size;

    const float* pos = (const float*)d_in[0];   // [4096,3] f32; cell/pbc unused

    // workspace: counts[4096] + offsets[4097]
    int* counts  = (int*)d_ws;
    int* offsets = counts + N_ATOMS;

    // output layout (return-order concat): idx_i[2P] idx_j[2P] Rij[2P,3]
    // cell_offset[2P,3] mask[2P] num_pairs[1]
    int*   idx_i  = (int*)d_out;
    int*   idx_j  = idx_i + TWO_P;
    float* rij    = (float*)(idx_j + TWO_P);
    int*   cell   = (int*)(rij + (size_t)3 * TWO_P);
    int*   mask   = cell + (size_t)3 * TWO_P;
    int*   npairs = mask + TWO_P;

    nl_count_wmma<<<NTILE / 8, 256, 0, stream>>>(pos, counts);

    nl_scan<<<1, 1024, 0, stream>>>(counts, offsets, npairs);

    nl_pad<<<(TWO_P + 255) / 256, 256, 0, stream>>>(idx_i, idx_j, rij, cell, mask);

    nl_emit_wmma<<<NTILE / 8, 256, 0, stream>>>(pos, offsets, idx_i, idx_j, rij, mask);
}